// BestHits_65635690218324
// MI455X (gfx1250) — compile-verified
//
#include <hip/hip_runtime.h>
#include <stdint.h>

// Problem constants (from reference: N=8192, G=32, S=256, TAU=0.1)
#define NN       8192
#define SBLK     256          // softmax block width == block tile size
#define NBLK     32           // NN / SBLK
#define NPAIRS   528          // NBLK*(NBLK+1)/2 triangular pairs
#define TILE     64           // pass-2 sub-tile edge
#define LSTRIDE  68           // LDS row stride in floats (64 + 4 pad, TDM pad feature)
#define K2       14.426950408889634f   // log2(e)/TAU : exp(x/tau) == exp2(x*K2)

typedef __attribute__((ext_vector_type(4))) unsigned int v4u;
typedef __attribute__((ext_vector_type(8))) int          v8i;
typedef __attribute__((ext_vector_type(4))) int          v4i;

__device__ __forceinline__ float fexp2(float x) { return __builtin_amdgcn_exp2f(x); }

// Issue one TDM 2D tile load: 64x64 fp32 tile, tensor row stride NN elements,
// destination LDS gets 4 dwords of padding after every 64 dwords (row stride 68).
// D# layout per CDNA5 ISA 8.3/8.4 (group0: count/lds_addr/global_addr/type,
// group1: data_size=4B, pad_enable, pad_interval=5 (64 dw), pad_amount=3 (4 dw),
// tensor_dim0/1 = NN, tile_dim0/1 = 64, tensor_dim0_stride = NN).
__device__ __forceinline__ void tdm_load_tile64(const float* gsrc, unsigned int lds_off) {
  unsigned long long ga = (unsigned long long)(uintptr_t)gsrc;
  v4u g0;
  g0.x = 1u;                                            // count=1 (valid user descriptor)
  g0.y = lds_off;                                       // lds_addr (bytes)
  g0.z = (unsigned int)ga;                              // global_addr[31:0]
  g0.w = (unsigned int)((ga >> 32) & 0x01FFFFFFull)     // global_addr[56:32]
         | (2u << 30);                                  // type=2 ("image")
  v8i g1;
  g1[0] = (int)((2u << 16)        // data_size = 4 bytes
              | (1u << 20)        // pad_enable
              | (5u << 22)        // pad_interval: 8<<5 = 256B = 64 dwords
              | (3u << 25));      // pad_amount: 3 -> 4 dwords (16B)
  g1[1] = (int)(((unsigned)NN & 0xFFFFu) << 16);                         // tensor_dim0 lo16
  g1[2] = (int)(((unsigned)NN >> 16) | (((unsigned)NN & 0xFFFFu) << 16));// dim0 hi | dim1 lo
  g1[3] = (int)(((unsigned)NN >> 16) | ((unsigned)TILE << 16));          // dim1 hi | tile_dim0
  g1[4] = (int)TILE;                                                     // tile_dim1 (tile_dim2=0)
  g1[5] = (int)NN;                                                       // tensor_dim0_stride lo32
  g1[6] = 0;                                                             // stride hi | dim1_stride lo
  g1[7] = 0;
  v4i gz4  = {0, 0, 0, 0};                    // group2 (unused, 2D tile)
  v8i gz8  = {0, 0, 0, 0, 0, 0, 0, 0};        // group3/extra (unused, zero-filled)
  // clang-23 / therock-10.0 form: 6 args (v4u, v8i, v4i, v4i, v8i, i32 cpol)
  __builtin_amdgcn_tensor_load_to_lds(g0, g1, gz4, gz4, gz8, 0);
}

__global__ __launch_bounds__(256)
void besthits_fused_kernel(const float* __restrict__ Sm, float* __restrict__ Out) {
  __shared__ float MA[SBLK], RLA[SBLK], MB[SBLK], RLB[SBLK];
  __shared__ float bufA[2][TILE * LSTRIDE];
  __shared__ float bufB[2][TILE * LSTRIDE];

  // Decode triangular pair index -> (bi <= bj)
  int p = (int)blockIdx.x;
  int bi = 0;
  while (p >= (NBLK - bi)) { p -= (NBLK - bi); ++bi; }
  const int bj = bi + p;
  const int RI = bi * SBLK, CJ = bj * SBLK;   // tile A = s[RI.., CJ..]
  const int RJ = bj * SBLK, CI = bi * SBLK;   // tile B = s[RJ.., CI..]

  const int tid  = (int)threadIdx.x;
  const int lane = tid & 31;
  const int wave = tid >> 5;

  // Prologue: kick off the first pair of TDM tile DMAs; overlaps with pass 1.
  if (wave == 0) {
    tdm_load_tile64(Sm + (size_t)RI * NN + CJ, (unsigned int)(uintptr_t)&bufA[0][0]);
    tdm_load_tile64(Sm + (size_t)RJ * NN + CI, (unsigned int)(uintptr_t)&bufB[0][0]);
  }

  // ---- Pass 1: per-row blockwise softmax stats (max, 1/sumexp), diag masked ----
  for (int t = 0; t < 2; ++t) {
    const int rowBase = (t == 0) ? RI : RJ;
    const int colBase = (t == 0) ? CJ : CI;
    float* Mdst = (t == 0) ? MA : MB;
    float* Ldst = (t == 0) ? RLA : RLB;
    for (int rr = wave; rr < SBLK; rr += 8) {
      const int i = rowBase + rr;
      const float* rowp = Sm + (size_t)i * NN + colBase;
      float4 x0 = *(const float4*)(rowp + 4 * lane);
      float4 x1 = *(const float4*)(rowp + 128 + 4 * lane);
      float v[8] = {x0.x, x0.y, x0.z, x0.w, x1.x, x1.y, x1.z, x1.w};
      const int dc = i - colBase;      // masked column (in [0,256) only on diag tiles)
      const int c0 = 4 * lane;
#pragma unroll
      for (int q = 0; q < 8; ++q) {
        int cc = (q < 4) ? (c0 + q) : (128 + c0 + q - 4);
        if (cc == dc) v[q] = -__builtin_inff();
      }
      float m = v[0];
#pragma unroll
      for (int q = 1; q < 8; ++q) m = fmaxf(m, v[q]);
#pragma unroll
      for (int off = 16; off > 0; off >>= 1) m = fmaxf(m, __shfl_xor(m, off, 32));
      float s = 0.f;
#pragma unroll
      for (int q = 0; q < 8; ++q) s += fexp2((v[q] - m) * K2);
#pragma unroll
      for (int off = 16; off > 0; off >>= 1) s += __shfl_xor(s, off, 32);
      if (lane == 0) { Mdst[rr] = m; Ldst[rr] = __builtin_amdgcn_rcpf(s); }
    }
  }
  __syncthreads();

  // ---- Pass 2: 16 sub-tile pairs, double-buffered TDM, write out[I,J] and out[J,I] ----
  const int c  = tid & 63;
  const int r0 = tid >> 6;

  for (int it = 0; it < 16; ++it) {
    const int u = it >> 2, v = it & 3;
    if (wave == 0) {
      if (it + 1 < 16) {
        const int u2 = (it + 1) >> 2, v2 = (it + 1) & 3, b2 = (it + 1) & 1;
        tdm_load_tile64(Sm + (size_t)(RI + u2 * TILE) * NN + CJ + v2 * TILE,
                        (unsigned int)(uintptr_t)&bufA[b2][0]);
        tdm_load_tile64(Sm + (size_t)(RJ + v2 * TILE) * NN + CI + u2 * TILE,
                        (unsigned int)(uintptr_t)&bufB[b2][0]);
        __builtin_amdgcn_s_wait_tensorcnt(2);   // current pair done (TDM in-order per wave)
      } else {
        __builtin_amdgcn_s_wait_tensorcnt(0);
      }
    }
    __syncthreads();   // current buffers visible to all waves

    float* A  = &bufA[it & 1][0];
    float* Bt = &bufB[it & 1][0];

    // Phase A: normalize in place (one exp per staged element), keep own values in regs.
    float ea[16], eb[16];
#pragma unroll 4
    for (int k = 0; k < 16; ++k) {
      const int r  = r0 + 4 * k;
      const int ra = u * TILE + r;
      const int rb = v * TILE + r;
      float va = fexp2((A[r * LSTRIDE + c]  - MA[ra]) * K2) * RLA[ra];
      float vb = fexp2((Bt[r * LSTRIDE + c] - MB[rb]) * K2) * RLB[rb];
      A[r * LSTRIDE + c]  = va;  ea[k] = va;
      Bt[r * LSTRIDE + c] = vb;  eb[k] = vb;
    }
    __syncthreads();

    // Phase B: reciprocal products; transposed factor comes from padded LDS.
#pragma unroll 4
    for (int k = 0; k < 16; ++k) {
      const int r   = r0 + 4 * k;
      float val  = ea[k] * Bt[c * LSTRIDE + r];
      float val2 = eb[k] * A[c * LSTRIDE + r];
      const int gi  = RI + u * TILE + r, gj  = CJ + v * TILE + c;
      const int gi2 = RJ + v * TILE + r, gj2 = CI + u * TILE + c;
      if (gi  == gj)  val  = 0.f;   // masked diagonal -> softmax value is 0
      if (gi2 == gj2) val2 = 0.f;
      Out[(size_t)gi  * NN + gj]  = val;
      Out[(size_t)gi2 * NN + gj2] = val2;
    }
    __syncthreads();   // safe to overwrite this buffer with it+2's DMA
  }
}

extern "C" void kernel_launch(void* const* d_in, const int* in_sizes, int n_in,
                              void* d_out, int out_size, void* d_ws, size_t ws_size,
                              hipStream_t stream) {
  (void)in_sizes; (void)n_in; (void)d_ws; (void)ws_size; (void)out_size;
  const float* Sm = (const float*)d_in[0];
  float* Out = (float*)d_out;
  besthits_fused_kernel<<<dim3(NPAIRS), dim3(256), 0, stream>>>(Sm, Out);
}